// EdgeMatcher_31421980737607
// MI455X (gfx1250) — compile-verified
//
#include <hip/hip_runtime.h>
#include <math.h>

#define REACH_INV (1.0f / 10.0f)
#define MAX_DIST_F 0.05f
#define WAVES_PER_BLOCK 8

// One wave32 per edge point. Lane L handles samples L, L+32, L+64, ...
// Within one loop iteration the 32 lanes cover 32 *consecutive* samples
// (~1.4 px apart on a 1024^2 image), so the wave's gathers are spatially
// dense and the +32-ahead prefetch streams the line corridor into WGP$.
__global__ __launch_bounds__(256) void edge_match_kernel(
    const float* __restrict__ pts,   // (N,2)
    const float* __restrict__ nrm,   // (N,2)
    const float* __restrict__ img,   // (H,W)  (edge_image; negated on the fly)
    const int*   __restrict__ pS,    // n_samples (device scalar)
    float* __restrict__ out,         // d_out: [loss][deltas(N,2) flat]
    float* __restrict__ ws_dist,     // N floats of masked distances
    int N, int H, int W)
{
    const int lane = threadIdx.x & 31;
    const int wave = threadIdx.x >> 5;
    const int n = blockIdx.x * WAVES_PER_BLOCK + wave;
    if (n >= N) return;           // uniform per wave

    const int S = *pS;
    const float maxf = 3.4028234663852886e38f;

    const float px = pts[2 * n + 0];
    const float py = pts[2 * n + 1];
    const float nx = nrm[2 * n + 0];
    const float ny = nrm[2 * n + 1];
    const float nlen = sqrtf(nx * nx + ny * ny);
    const float dx = nx / nlen;
    const float dy = ny / nlen;

    // Line/box parametric intersection, exactly as the reference (no swap on sign).
    const bool zx = fabsf(dx) <= 0.001f;
    const bool zy = fabsf(dy) <= 0.001f;
    const float sdx = zx ? 1.0f : dx;
    const float sdy = zy ? 1.0f : dy;
    const float t_left   = zx ? -maxf : (0.0f - px) / sdx;
    const float t_right  = zx ?  maxf : ((float)(W - 1) - px) / sdx;
    const float t_top    = zy ? -maxf : (0.0f - py) / sdy;
    const float t_bottom = zy ?  maxf : ((float)(H - 1) - py) / sdy;
    const float t_min = fmaxf(t_left, t_top);
    const float t_max = fminf(t_right, t_bottom);

    const float p1x = px + t_min * dx, p1y = py + t_min * dy;
    const float p2x = px + t_max * dx, p2y = py + t_max * dy;
    const float vx = p2x - p1x, vy = p2y - p1y;
    const float step = (S > 1) ? (1.0f / (float)(S - 1)) : 0.0f;
    const float Wm1 = (float)(W - 1), Hm1 = (float)(H - 1);

    // Running argmins (first-occurrence tie-break: strict < in-lane since i
    // increases, index compare cross-lane).
    float bestAdj  = maxf; int bestAdjIdx  = 0x7fffffff;   // argmin of plain adjusted
    float bestDist = maxf; int bestDistIdx = 0x7fffffff;   // argmin of dist
    float bestAdjPrime = maxf;                             // adjusted - 1e-6*e at dist-argmin

    for (int i = lane; i < S; i += 32) {
        // Branchless software prefetch of the next iteration's sample:
        // both bilinear rows (they live in different cachelines).
        {
            const int ip = min(i + 32, S - 1);
            const float tp  = (float)ip * step;
            const float sxp = fminf(fmaxf(fmaf(tp, vx, p1x), 0.0f), Wm1);
            const float syp = fminf(fmaxf(fmaf(tp, vy, p1y), 0.0f), Hm1);
            const int ixp = (int)sxp;
            const int iyp = (int)syp;
            const int iyp1 = min(iyp + 1, H - 1);
            __builtin_prefetch(&img[iyp  * W + ixp], 0, 3);  // global_prefetch_b8
            __builtin_prefetch(&img[iyp1 * W + ixp], 0, 3);
        }

        const float t  = (float)i * step;
        const float sx = fminf(fmaxf(fmaf(t, vx, p1x), 0.0f), Wm1);
        const float sy = fminf(fmaxf(fmaf(t, vy, p1y), 0.0f), Hm1);
        const float x0 = floorf(sx), y0 = floorf(sy);
        const float wx = sx - x0,    wy = sy - y0;
        const int ix0 = (int)x0, iy0 = (int)y0;
        const int ix1 = min(ix0 + 1, W - 1);
        const int iy1 = min(iy0 + 1, H - 1);

        // 4 independent gathers; shared row bases keep address VALU low.
        const float* row0 = img + (size_t)iy0 * W;
        const float* row1 = img + (size_t)iy1 * W;
        const float v00 = row0[ix0];
        const float v01 = row0[ix1];
        const float v10 = row1[ix0];
        const float v11 = row1[ix1];

        const float omwx = 1.0f - wx, omwy = 1.0f - wy;
        const float val = -(v00 * omwx * omwy + v01 * wx * omwy +
                            v10 * omwx * wy   + v11 * wx * wy);

        const float ddx = sx - px, ddy = sy - py;
        const float dist = sqrtf(ddx * ddx + ddy * ddy);
        const float e = __expf(-dist * REACH_INV);
        const float adj = val * e;

        if (adj < bestAdj)  { bestAdj = adj; bestAdjIdx = i; }
        if (dist < bestDist) {
            bestDist = dist; bestDistIdx = i;
            bestAdjPrime = adj - 1e-6f * e;
        }
    }

    // wave32 butterfly argmin (total order: value, then index) — all lanes converge.
    #pragma unroll
    for (int off = 16; off > 0; off >>= 1) {
        const float oA  = __shfl_xor(bestAdj, off, 32);
        const int   oAi = __shfl_xor(bestAdjIdx, off, 32);
        if (oA < bestAdj || (oA == bestAdj && oAi < bestAdjIdx)) {
            bestAdj = oA; bestAdjIdx = oAi;
        }
        const float oD  = __shfl_xor(bestDist, off, 32);
        const int   oDi = __shfl_xor(bestDistIdx, off, 32);
        const float oP  = __shfl_xor(bestAdjPrime, off, 32);
        if (oD < bestDist || (oD == bestDist && oDi < bestDistIdx)) {
            bestDist = oD; bestDistIdx = oDi; bestAdjPrime = oP;
        }
    }

    // The epsilon nudge strictly lowers the ref sample; resolve the final
    // argmin with JAX first-occurrence tie-breaking.
    const bool useRef = (bestAdjPrime < bestAdj) ||
                        (bestAdjPrime == bestAdj && bestDistIdx < bestAdjIdx);
    const int idx = useRef ? bestDistIdx : bestAdjIdx;

    if (lane == 0) {
        const float t  = (float)idx * step;
        const float sx = fminf(fmaxf(fmaf(t, vx, p1x), 0.0f), Wm1);
        const float sy = fminf(fmaxf(fmaf(t, vy, p1y), 0.0f), Hm1);
        const float dX = sx - px, dY = sy - py;
        const float rX = dX / (float)H * 2.0f;
        const float rY = dY / (float)H * 2.0f;
        const float distRel = sqrtf(rX * rX + rY * rY);
        const bool mask = distRel < MAX_DIST_F;
        out[1 + 2 * n + 0] = mask ? dX : 0.0f;
        out[1 + 2 * n + 1] = mask ? dY : 0.0f;
        ws_dist[n] = mask ? distRel : 0.0f;
    }
}

// Deterministic fixed-tree mean reduction (no float atomics; d_out[0] written
// unconditionally every call, so 0xAA poison never survives).
__global__ __launch_bounds__(256) void loss_reduce_kernel(
    const float* __restrict__ ws_dist, float* __restrict__ out, int N)
{
    __shared__ float sdata[256];
    float s = 0.0f;
    for (int i = threadIdx.x; i < N; i += 256) s += ws_dist[i];
    sdata[threadIdx.x] = s;
    __syncthreads();
    for (int o = 128; o > 0; o >>= 1) {
        if ((int)threadIdx.x < o) sdata[threadIdx.x] += sdata[threadIdx.x + o];
        __syncthreads();
    }
    if (threadIdx.x == 0) out[0] = sdata[0] / (float)N;
}

extern "C" void kernel_launch(void* const* d_in, const int* in_sizes, int n_in,
                              void* d_out, int out_size, void* d_ws, size_t ws_size,
                              hipStream_t stream) {
    const float* pts = (const float*)d_in[0];   // edge_points (N,2) f32
    const float* nrm = (const float*)d_in[1];   // edge_normals (N,2) f32
    const float* img = (const float*)d_in[2];   // edge_image (H,W) f32
    const int*   pS  = (const int*)d_in[3];     // n_samples scalar

    const int N  = in_sizes[0] / 2;
    const int HW = in_sizes[2];
    int W = (int)(sqrt((double)HW) + 0.5);      // square image (1024x1024)
    int H = HW / W;

    float* out = (float*)d_out;                 // [loss][deltas flat]
    float* ws  = (float*)d_ws;                  // N floats

    const int blocks = (N + WAVES_PER_BLOCK - 1) / WAVES_PER_BLOCK;
    edge_match_kernel<<<blocks, 32 * WAVES_PER_BLOCK, 0, stream>>>(
        pts, nrm, img, pS, out, ws, N, H, W);
    loss_reduce_kernel<<<1, 256, 0, stream>>>(ws, out, N);
}